// GAT_88192858456221
// MI455X (gfx1250) — compile-verified
//
#include <hip/hip_runtime.h>
#include <math.h>

// Problem constants (from the reference)
#define NIN   128   // IN_F
#define D1    128   // HEADS*HID
#define HIDC  32
#define NHEAD 4
#define OUTF  40
#define LN_EPS 1e-5f
#define INT_MIN_S ((int)0x80000000)

typedef __attribute__((ext_vector_type(2))) float v2f;
typedef __attribute__((ext_vector_type(8))) float v8f;

// ---------------- monotonic float<->int mapping for atomic segment-max ----
__device__ __forceinline__ int map_f(float f) {
    int i = __float_as_int(f);
    return i >= 0 ? i : (i ^ 0x7FFFFFFF);
}
__device__ __forceinline__ float unmap_f(int i) {
    return __int_as_float(i >= 0 ? i : (i ^ 0x7FFFFFFF));
}

// ---------------- init: fold biases into accumulators, reset max/sum ------
__global__ void gat_init_kernel(float* __restrict__ out1, const float* __restrict__ b1,
                                int* __restrict__ mbuf, float* __restrict__ sbuf,
                                int* __restrict__ m2, float* __restrict__ s2,
                                float* __restrict__ out, const float* __restrict__ b2,
                                int n_nodes)
{
    long long i = (long long)blockIdx.x * blockDim.x + threadIdx.x;
    long long nd1 = (long long)n_nodes * D1;
    if (i < nd1)                       out1[i] = b1[i & (D1 - 1)];
    if (i < (long long)n_nodes*NHEAD) { mbuf[i] = INT_MIN_S; sbuf[i] = 0.f; }
    if (i < (long long)n_nodes)       { m2[i]   = INT_MIN_S; s2[i]   = 0.f; }
    if (i < (long long)n_nodes*OUTF)   out[i]  = b2[i % OUTF];
}

// ---------------- layer-1 GEMMs: xl1 = x@Wl1+bl1, xr1 = x@Wr1+br1 ---------
// fp32 WMMA 16x16x4. Block = 512 threads = 16 waves; one 16-row tile/block.
// Waves 0..7 -> Wl1 column tiles, waves 8..15 -> Wr1 column tiles.
__global__ __launch_bounds__(512)
void gat_gemm1_kernel(const float* __restrict__ x,
                      const float* __restrict__ Wl, const float* __restrict__ bl,
                      const float* __restrict__ Wr, const float* __restrict__ br,
                      float* __restrict__ xl, float* __restrict__ xr, int n_nodes)
{
    __shared__ float xs[16 * NIN];
    int m0 = blockIdx.x * 16;
    for (int i = threadIdx.x; i < 16 * NIN; i += 512) {
        int r = i >> 7, c = i & (NIN - 1);
        int row = m0 + r; if (row >= n_nodes) row = n_nodes - 1;
        xs[i] = x[(size_t)row * NIN + c];
    }
    __syncthreads();

    int wave = threadIdx.x >> 5;
    int lane = threadIdx.x & 31;
    int half = lane >> 4;        // K sub-pair / C row-half selector
    int l16  = lane & 15;

    const float* W   = (wave < 8) ? Wl : Wr;
    const float* bb  = (wave < 8) ? bl : br;
    float*       dst = (wave < 8) ? xl : xr;
    int n0 = (wave & 7) * 16;

    v8f acc = {};
    for (int k0 = 0; k0 < NIN; k0 += 4) {
        v2f a, b;
        // A 16x4 (ISA 7.12.2): lane row = l16, VGPR0/1 = K = k0+2*half+{0,1}
        a.x = xs[l16 * NIN + k0 + 2 * half + 0];
        a.y = xs[l16 * NIN + k0 + 2 * half + 1];
        // B 4x16: lane col = l16, VGPR r holds rows (k0+2*half+r)
        b.x = W[(size_t)(k0 + 2 * half + 0) * D1 + n0 + l16];
        b.y = W[(size_t)(k0 + 2 * half + 1) * D1 + n0 + l16];
        acc = __builtin_amdgcn_wmma_f32_16x16x4_f32(false, a, false, b,
                                                    (short)0, acc, false, false);
    }
    float bias = bb[n0 + l16];
    #pragma unroll
    for (int r = 0; r < 8; ++r) {
        int row = m0 + r + 8 * half;   // C/D layout: VGPR r = rows r, r+8
        if (row < n_nodes)
            dst[(size_t)row * D1 + n0 + l16] = acc[r] + bias;
    }
}

// ---------------- layer-1 edge score + segment max ------------------------
__global__ void gat_edge1_score_kernel(const long long* __restrict__ ei, long long E,
                                       const float* __restrict__ xl,
                                       const float* __restrict__ xr,
                                       const float* __restrict__ att,
                                       float* __restrict__ ebuf, int* __restrict__ mbuf)
{
    long long idx = (long long)blockIdx.x * blockDim.x + threadIdx.x;
    if (idx >= E * NHEAD) return;
    long long e = idx >> 2;
    int h = (int)(idx & 3);
    long long src = ei[e], tgt = ei[E + e];
    const float* pl = xl + src * D1 + h * HIDC;
    const float* pr = xr + tgt * D1 + h * HIDC;
    const float* pa = att + h * HIDC;
    float s = 0.f;
    #pragma unroll
    for (int c = 0; c < HIDC; ++c) {
        float v = pl[c] + pr[c];
        v = v > 0.f ? v : 0.2f * v;          // leaky_relu(0.2)
        s += v * pa[c];
    }
    ebuf[idx] = s;
    atomicMax(&mbuf[tgt * NHEAD + h], map_f(s));
}

// ---------------- layer-1 exp(e-m) + segment sum --------------------------
__global__ void gat_edge1_exp_kernel(const long long* __restrict__ ei, long long E,
                                     float* __restrict__ ebuf,
                                     const int* __restrict__ mbuf,
                                     float* __restrict__ sbuf)
{
    long long idx = (long long)blockIdx.x * blockDim.x + threadIdx.x;
    if (idx >= E * NHEAD) return;
    long long e = idx >> 2;
    int h = (int)(idx & 3);
    long long tgt = ei[E + e];
    int mi = mbuf[tgt * NHEAD + h];
    float m = (mi == INT_MIN_S) ? 0.f : unmap_f(mi);
    float p = expf(ebuf[idx] - m);
    ebuf[idx] = p;
    atomicAdd(&sbuf[tgt * NHEAD + h], p);
}

// ---------------- layer-1 weighted aggregation ----------------------------
__global__ void gat_edge1_agg_kernel(const long long* __restrict__ ei, long long E,
                                     const float* __restrict__ xl,
                                     const float* __restrict__ pbuf,
                                     const float* __restrict__ sbuf,
                                     float* __restrict__ out1)
{
    long long idx = (long long)blockIdx.x * blockDim.x + threadIdx.x;
    if (idx >= E * D1) return;
    long long e = idx >> 7;
    int cc = (int)(idx & (D1 - 1));
    int h = cc >> 5;
    long long src = ei[e], tgt = ei[E + e];
    float alpha = pbuf[e * NHEAD + h] / (sbuf[tgt * NHEAD + h] + 1e-16f);
    atomicAdd(&out1[tgt * D1 + cc], alpha * xl[src * D1 + cc]);
}

// ---------------- LayerNorm + ELU, in place over out1 ---------------------
__global__ __launch_bounds__(128)
void gat_ln_elu_kernel(float* __restrict__ h, const float* __restrict__ g,
                       const float* __restrict__ be)
{
    __shared__ float red[D1];
    int n = blockIdx.x, t = threadIdx.x;
    float v = h[(size_t)n * D1 + t];
    red[t] = v;
    __syncthreads();
    for (int s = 64; s > 0; s >>= 1) { if (t < s) red[t] += red[t + s]; __syncthreads(); }
    float mu = red[0] * (1.0f / D1);
    __syncthreads();
    float d = v - mu;
    red[t] = d * d;
    __syncthreads();
    for (int s = 64; s > 0; s >>= 1) { if (t < s) red[t] += red[t + s]; __syncthreads(); }
    float var = red[0] * (1.0f / D1);
    float y = d * rsqrtf(var + LN_EPS) * g[t] + be[t];
    h[(size_t)n * D1 + t] = y > 0.f ? y : expm1f(y);   // ELU(alpha=1)
}

// ---------------- layer-2 GEMMs: xl2/xr2 = h@W2+b2 (128 -> 40) ------------
// Block = 192 threads = 6 waves; waves 0..2 -> Wl2 tiles, 3..5 -> Wr2 tiles.
__global__ __launch_bounds__(192)
void gat_gemm2_kernel(const float* __restrict__ h,
                      const float* __restrict__ Wl, const float* __restrict__ bl,
                      const float* __restrict__ Wr, const float* __restrict__ br,
                      float* __restrict__ xl2, float* __restrict__ xr2, int n_nodes)
{
    __shared__ float hs[16 * D1];
    int m0 = blockIdx.x * 16;
    for (int i = threadIdx.x; i < 16 * D1; i += 192) {
        int r = i >> 7, c = i & (D1 - 1);
        int row = m0 + r; if (row >= n_nodes) row = n_nodes - 1;
        hs[i] = h[(size_t)row * D1 + c];
    }
    __syncthreads();

    int wave = threadIdx.x >> 5;
    int lane = threadIdx.x & 31;
    int half = lane >> 4, l16 = lane & 15;
    const float* W   = (wave < 3) ? Wl : Wr;
    const float* bb  = (wave < 3) ? bl : br;
    float*       dst = (wave < 3) ? xl2 : xr2;
    int n0 = (wave % 3) * 16;
    int col  = n0 + l16;
    int colc = col < OUTF ? col : (OUTF - 1);
    float cmask = col < OUTF ? 1.0f : 0.0f;   // zero-pad partial B tile

    v8f acc = {};
    for (int k0 = 0; k0 < D1; k0 += 4) {
        v2f a, b;
        a.x = hs[l16 * D1 + k0 + 2 * half + 0];
        a.y = hs[l16 * D1 + k0 + 2 * half + 1];
        b.x = cmask * W[(size_t)(k0 + 2 * half + 0) * OUTF + colc];
        b.y = cmask * W[(size_t)(k0 + 2 * half + 1) * OUTF + colc];
        acc = __builtin_amdgcn_wmma_f32_16x16x4_f32(false, a, false, b,
                                                    (short)0, acc, false, false);
    }
    if (col < OUTF) {
        float bias = bb[col];
        #pragma unroll
        for (int r = 0; r < 8; ++r) {
            int row = m0 + r + 8 * half;
            if (row < n_nodes)
                dst[(size_t)row * OUTF + col] = acc[r] + bias;
        }
    }
}

// ---------------- layer-2 edge score + segment max (heads=1, C=40) --------
__global__ void gat_edge2_score_kernel(const long long* __restrict__ ei, long long E,
                                       const float* __restrict__ xl2,
                                       const float* __restrict__ xr2,
                                       const float* __restrict__ att2,
                                       float* __restrict__ e2, int* __restrict__ m2)
{
    long long e = (long long)blockIdx.x * blockDim.x + threadIdx.x;
    if (e >= E) return;
    long long src = ei[e], tgt = ei[E + e];
    const float* pl = xl2 + src * OUTF;
    const float* pr = xr2 + tgt * OUTF;
    float s = 0.f;
    #pragma unroll
    for (int c = 0; c < OUTF; ++c) {
        float v = pl[c] + pr[c];
        v = v > 0.f ? v : 0.2f * v;
        s += v * att2[c];
    }
    e2[e] = s;
    atomicMax(&m2[tgt], map_f(s));
}

__global__ void gat_edge2_exp_kernel(const long long* __restrict__ ei, long long E,
                                     float* __restrict__ e2,
                                     const int* __restrict__ m2,
                                     float* __restrict__ s2)
{
    long long e = (long long)blockIdx.x * blockDim.x + threadIdx.x;
    if (e >= E) return;
    long long tgt = ei[E + e];
    int mi = m2[tgt];
    float m = (mi == INT_MIN_S) ? 0.f : unmap_f(mi);
    float p = expf(e2[e] - m);
    e2[e] = p;
    atomicAdd(&s2[tgt], p);
}

__global__ void gat_edge2_agg_kernel(const long long* __restrict__ ei, long long E,
                                     const float* __restrict__ xl2,
                                     const float* __restrict__ p2,
                                     const float* __restrict__ s2,
                                     float* __restrict__ out)
{
    long long idx = (long long)blockIdx.x * blockDim.x + threadIdx.x;
    if (idx >= E * OUTF) return;
    long long e = idx / OUTF;
    int c = (int)(idx - e * OUTF);
    long long src = ei[e], tgt = ei[E + e];
    float alpha = p2[e] / (s2[tgt] + 1e-16f);
    atomicAdd(&out[tgt * OUTF + c], alpha * xl2[src * OUTF + c]);
}

// --------------------------------------------------------------------------
extern "C" void kernel_launch(void* const* d_in, const int* in_sizes, int n_in,
                              void* d_out, int out_size, void* d_ws, size_t ws_size,
                              hipStream_t stream)
{
    const float*     x    = (const float*)d_in[0];
    const long long* ei   = (const long long*)d_in[1];   // int64 per reference
    const float*     Wl1  = (const float*)d_in[2];
    const float*     bl1  = (const float*)d_in[3];
    const float*     Wr1  = (const float*)d_in[4];
    const float*     br1  = (const float*)d_in[5];
    const float*     att1 = (const float*)d_in[6];
    const float*     b1   = (const float*)d_in[7];
    const float*     g1   = (const float*)d_in[8];
    const float*     be1  = (const float*)d_in[9];
    const float*     Wl2  = (const float*)d_in[10];
    const float*     bl2  = (const float*)d_in[11];
    const float*     Wr2  = (const float*)d_in[12];
    const float*     br2  = (const float*)d_in[13];
    const float*     att2 = (const float*)d_in[14];
    const float*     b2   = (const float*)d_in[15];
    float* out = (float*)d_out;

    const int       n_nodes = in_sizes[0] / NIN;       // 50000
    const long long E       = in_sizes[1] / 2;         // 800000

    // -------- workspace carve-out (~111 MB) -------------------------------
    char* ws = (char*)d_ws;
    size_t off = 0;
    auto carve = [&](size_t bytes) -> void* {
        void* p = ws + off;
        off = (off + bytes + 255) & ~(size_t)255;
        return p;
    };
    float* xl1  = (float*)carve((size_t)n_nodes * D1 * sizeof(float));
    float* xr1  = (float*)carve((size_t)n_nodes * D1 * sizeof(float));
    float* h1   = (float*)carve((size_t)n_nodes * D1 * sizeof(float)); // accum -> LN/ELU in place
    float* ebuf = (float*)carve((size_t)E * NHEAD * sizeof(float));    // e, then p
    int*   mbuf = (int*)  carve((size_t)n_nodes * NHEAD * sizeof(int));
    float* sbuf = (float*)carve((size_t)n_nodes * NHEAD * sizeof(float));
    float* xl2  = (float*)carve((size_t)n_nodes * OUTF * sizeof(float));
    float* xr2  = (float*)carve((size_t)n_nodes * OUTF * sizeof(float));
    float* e2   = (float*)carve((size_t)E * sizeof(float));
    int*   m2   = (int*)  carve((size_t)n_nodes * sizeof(int));
    float* s2   = (float*)carve((size_t)n_nodes * sizeof(float));
    (void)ws_size; (void)n_in; (void)out_size;

    const int BT = 256;
    long long t;

    // 1) init accumulators (bias folded in) and segment max/sum buffers
    t = (long long)n_nodes * D1;
    gat_init_kernel<<<(unsigned)((t + BT - 1) / BT), BT, 0, stream>>>(
        h1, b1, mbuf, sbuf, m2, s2, out, b2, n_nodes);

    // 2) layer-1 GEMMs (fp32 WMMA)
    gat_gemm1_kernel<<<(unsigned)((n_nodes + 15) / 16), 512, 0, stream>>>(
        x, Wl1, bl1, Wr1, br1, xl1, xr1, n_nodes);

    // 3) layer-1 attention scores + segment max
    t = E * NHEAD;
    gat_edge1_score_kernel<<<(unsigned)((t + BT - 1) / BT), BT, 0, stream>>>(
        ei, E, xl1, xr1, att1, ebuf, mbuf);

    // 4) exp(e-m) + segment sum
    gat_edge1_exp_kernel<<<(unsigned)((t + BT - 1) / BT), BT, 0, stream>>>(
        ei, E, ebuf, mbuf, sbuf);

    // 5) weighted aggregation
    t = E * D1;
    gat_edge1_agg_kernel<<<(unsigned)((t + BT - 1) / BT), BT, 0, stream>>>(
        ei, E, xl1, ebuf, sbuf, h1);

    // 6) LayerNorm + ELU in place
    gat_ln_elu_kernel<<<(unsigned)n_nodes, 128, 0, stream>>>(h1, g1, be1);

    // 7) layer-2 GEMMs (fp32 WMMA, 40-col partial tiles)
    gat_gemm2_kernel<<<(unsigned)((n_nodes + 15) / 16), 192, 0, stream>>>(
        h1, Wl2, bl2, Wr2, br2, xl2, xr2, n_nodes);

    // 8) layer-2 scores + segment max
    gat_edge2_score_kernel<<<(unsigned)((E + BT - 1) / BT), BT, 0, stream>>>(
        ei, E, xl2, xr2, att2, e2, m2);

    // 9) exp + segment sum
    gat_edge2_exp_kernel<<<(unsigned)((E + BT - 1) / BT), BT, 0, stream>>>(
        ei, E, e2, m2, s2);

    // 10) aggregation into d_out (pre-initialized with b2)
    t = E * OUTF;
    gat_edge2_agg_kernel<<<(unsigned)((t + BT - 1) / BT), BT, 0, stream>>>(
        ei, E, xl2, e2, s2, out);
}